// Attention_31696858644692
// MI455X (gfx1250) — compile-verified
//
#include <hip/hip_runtime.h>
#include <hip/hip_bf16.h>

typedef __attribute__((ext_vector_type(2))) float v2f;
typedef __attribute__((ext_vector_type(8))) float v8f;

#define B_   8
#define C_   256
#define L_   4096
#define OC_  128
#define KC   32
#define LDS_STRIDE 36      // [row][k] tiles: 144B rows -> 16B aligned, bank-friendly
#define BSTR 132           // [k][n] tiles: 528B rows -> 16B aligned
#define BN_EPS 1e-5f

// ---------------- CDNA5 async global->LDS copy (ASYNCcnt-tracked) -----------
// VDST operand = wave-relative LDS byte offset (= low 32 bits of the generic
// pointer per the LDS aperture rule); address = 64-bit VGPR pair.
__device__ __forceinline__ void async_cp16(const float* g, const float* l) {
  asm volatile("global_load_async_to_lds_b128 %0, %1, off"
               :: "v"((unsigned int)(unsigned long long)l),
                  "v"((unsigned long long)g)
               : "memory");
}
__device__ __forceinline__ void wait_stage() {
  asm volatile("s_wait_asynccnt 0x0" ::: "memory");
  __syncthreads();
}

// ---------------- WMMA helper: exact f32 matrix op --------------------------
__device__ __forceinline__ v8f wmma4(v2f a, v2f b, v8f c) {
  return __builtin_amdgcn_wmma_f32_16x16x4_f32(false, a, false, b, (short)0, c,
                                               false, false);
}

// Wave computes 32(M) x 64(N) of the block tile. As[m][k], Bs[n][k] (k in regs).
__device__ __forceinline__ void wave_tile_nB(const float* As, const float* Bs,
                                             int wm, int wn, int half, int r,
                                             v8f (&acc)[2][4]) {
  #pragma unroll
  for (int k4 = 0; k4 < KC; k4 += 4) {
    const int kk = k4 + 2 * half;
    v2f a0 = *(const v2f*)&As[(wm * 32      + r) * LDS_STRIDE + kk];
    v2f a1 = *(const v2f*)&As[(wm * 32 + 16 + r) * LDS_STRIDE + kk];
    v2f b0 = *(const v2f*)&Bs[(wn * 64      + r) * LDS_STRIDE + kk];
    v2f b1 = *(const v2f*)&Bs[(wn * 64 + 16 + r) * LDS_STRIDE + kk];
    v2f b2 = *(const v2f*)&Bs[(wn * 64 + 32 + r) * LDS_STRIDE + kk];
    v2f b3 = *(const v2f*)&Bs[(wn * 64 + 48 + r) * LDS_STRIDE + kk];
    acc[0][0] = wmma4(a0, b0, acc[0][0]);
    acc[0][1] = wmma4(a0, b1, acc[0][1]);
    acc[0][2] = wmma4(a0, b2, acc[0][2]);
    acc[0][3] = wmma4(a0, b3, acc[0][3]);
    acc[1][0] = wmma4(a1, b0, acc[1][0]);
    acc[1][1] = wmma4(a1, b1, acc[1][1]);
    acc[1][2] = wmma4(a1, b2, acc[1][2]);
    acc[1][3] = wmma4(a1, b3, acc[1][3]);
  }
}

// Same, but B tile is k-major: Bs[k][n] with stride BSTR.
__device__ __forceinline__ void wave_tile_kB(const float* As, const float* Bs,
                                             int wm, int wn, int half, int r,
                                             v8f (&acc)[2][4]) {
  #pragma unroll
  for (int k4 = 0; k4 < KC; k4 += 4) {
    const int kk = k4 + 2 * half;
    v2f a0 = *(const v2f*)&As[(wm * 32      + r) * LDS_STRIDE + kk];
    v2f a1 = *(const v2f*)&As[(wm * 32 + 16 + r) * LDS_STRIDE + kk];
    const float* brow0 = &Bs[kk * BSTR + wn * 64 + r];
    const float* brow1 = brow0 + BSTR;
    v2f b0 = { brow0[0],  brow1[0]  };
    v2f b1 = { brow0[16], brow1[16] };
    v2f b2 = { brow0[32], brow1[32] };
    v2f b3 = { brow0[48], brow1[48] };
    acc[0][0] = wmma4(a0, b0, acc[0][0]);
    acc[0][1] = wmma4(a0, b1, acc[0][1]);
    acc[0][2] = wmma4(a0, b2, acc[0][2]);
    acc[0][3] = wmma4(a0, b3, acc[0][3]);
    acc[1][0] = wmma4(a1, b0, acc[1][0]);
    acc[1][1] = wmma4(a1, b1, acc[1][1]);
    acc[1][2] = wmma4(a1, b2, acc[1][2]);
    acc[1][3] = wmma4(a1, b3, acc[1][3]);
  }
}

// Async-stage a 128(row) x KC tile, row-major-in-k (stride srcStride) -> [row][k].
__device__ __forceinline__ void stage_A(const float* src, size_t srcStride,
                                        float* dst, int tid) {
  #pragma unroll
  for (int i = 0; i < 4; ++i) {
    const int q = tid + i * 256, row = q >> 3, c4 = (q & 7) << 2;
    async_cp16(src + (size_t)row * srcStride + c4, dst + row * LDS_STRIDE + c4);
  }
}

// Async-stage a KC(k) x 128(n) tile, row-major-in-n (stride srcStride) -> [k][n].
__device__ __forceinline__ void stage_Bk(const float* src, size_t srcStride,
                                         float* dst, int tid) {
  #pragma unroll
  for (int i = 0; i < 4; ++i) {
    const int q = tid + i * 256, krow = q >> 5, c4 = (q & 31) << 2;
    async_cp16(src + (size_t)krow * srcStride + c4, dst + krow * BSTR + c4);
  }
}

// ---------------- Kernel 1: fused g/theta/phi projections -------------------
__global__ __launch_bounds__(256) void k_proj(
    const float* __restrict__ x,
    const float* __restrict__ g_w,  const float* __restrict__ g_b,
    const float* __restrict__ th_w, const float* __restrict__ th_b,
    const float* __restrict__ ph_w, const float* __restrict__ ph_b,
    float* __restrict__ P) {
  __shared__ float As[2][128 * LDS_STRIDE];
  __shared__ float Bs[2][KC * BSTR];
  const int tid = threadIdx.x;
  const int b = blockIdx.z, mw = blockIdx.y;
  const size_t n0 = (size_t)blockIdx.x * 128;
  const float* W    = (mw == 0) ? g_w : (mw == 1 ? th_w : ph_w);
  const float* bias = (mw == 0) ? g_b : (mw == 1 ? th_b : ph_b);
  const float* xb = x + (size_t)b * C_ * L_ + n0;

  v8f acc[2][4] = {};
  const int wave = tid >> 5, lane = tid & 31;
  const int wm = wave >> 1, wn = wave & 1, half = lane >> 4, r = lane & 15;

  stage_A(W, C_, As[0], tid);
  stage_Bk(xb, L_, Bs[0], tid);
  wait_stage();
  int p = 0;
  for (int kc = 0; kc < C_; kc += KC, p ^= 1) {
    if (kc + KC < C_) {
      stage_A(W + kc + KC, C_, As[p ^ 1], tid);
      stage_Bk(xb + (size_t)(kc + KC) * L_, L_, Bs[p ^ 1], tid);
    }
    wave_tile_kB(As[p], Bs[p], wm, wn, half, r, acc);
    wait_stage();
  }

  float* Pout = P + ((size_t)b * 384 + (size_t)mw * 128) * L_ + n0;
  #pragma unroll
  for (int mi = 0; mi < 2; ++mi)
    #pragma unroll
    for (int v = 0; v < 8; ++v) {
      const int row = wm * 32 + mi * 16 + v + 8 * half;
      const float bv = bias[row];
      #pragma unroll
      for (int ni = 0; ni < 4; ++ni)
        Pout[(size_t)row * L_ + wn * 64 + ni * 16 + r] = acc[mi][ni][v] + bv;
    }
}

// ---------------- Kernel 2: S[b] = (1/L) * g_x @ phi_x^T (128x128) ----------
__global__ __launch_bounds__(256) void k_S(const float* __restrict__ P,
                                           float* __restrict__ S) {
  __shared__ float As[2][128 * LDS_STRIDE];
  __shared__ float Bs[2][128 * LDS_STRIDE];
  const int tid = threadIdx.x;
  const int b = blockIdx.y;
  const size_t k0 = (size_t)blockIdx.x * 256;            // K-split over L
  const float* G  = P + (size_t)b * 384 * L_ + k0;
  const float* PH = P + ((size_t)b * 384 + 256) * L_ + k0;

  v8f acc[2][4] = {};
  const int wave = tid >> 5, lane = tid & 31;
  const int wm = wave >> 1, wn = wave & 1, half = lane >> 4, r = lane & 15;

  stage_A(G, L_, As[0], tid);
  stage_A(PH, L_, Bs[0], tid);
  wait_stage();
  int p = 0;
  for (int kc = 0; kc < 256; kc += KC, p ^= 1) {
    if (kc + KC < 256) {
      stage_A(G + kc + KC, L_, As[p ^ 1], tid);
      stage_A(PH + kc + KC, L_, Bs[p ^ 1], tid);
    }
    wave_tile_nB(As[p], Bs[p], wm, wn, half, r, acc);
    wait_stage();
  }

  float* Sb = S + (size_t)b * 128 * 128;
  const float inv = 1.0f / (float)L_;
  #pragma unroll
  for (int mi = 0; mi < 2; ++mi)
    #pragma unroll
    for (int v = 0; v < 8; ++v) {
      const int row = wm * 32 + mi * 16 + v + 8 * half;
      #pragma unroll
      for (int ni = 0; ni < 4; ++ni)
        atomicAdd(&Sb[row * 128 + wn * 64 + ni * 16 + r], acc[mi][ni][v] * inv);
    }
}

// ---------------- Kernel 3: y[b] = S[b] @ theta_x[b] (128 x L) --------------
__global__ __launch_bounds__(256) void k_Y(const float* __restrict__ P,
                                           const float* __restrict__ S,
                                           float* __restrict__ Y) {
  __shared__ float As[2][128 * LDS_STRIDE];
  __shared__ float Bs[2][KC * BSTR];
  const int tid = threadIdx.x;
  const int b = blockIdx.y;
  const size_t n0 = (size_t)blockIdx.x * 128;
  const float* Sb = S + (size_t)b * 128 * 128;
  const float* TH = P + ((size_t)b * 384 + 128) * L_ + n0;

  v8f acc[2][4] = {};
  const int wave = tid >> 5, lane = tid & 31;
  const int wm = wave >> 1, wn = wave & 1, half = lane >> 4, r = lane & 15;

  stage_A(Sb, 128, As[0], tid);
  stage_Bk(TH, L_, Bs[0], tid);
  wait_stage();
  int p = 0;
  for (int kc = 0; kc < OC_; kc += KC, p ^= 1) {
    if (kc + KC < OC_) {
      stage_A(Sb + kc + KC, 128, As[p ^ 1], tid);
      stage_Bk(TH + (size_t)(kc + KC) * L_, L_, Bs[p ^ 1], tid);
    }
    wave_tile_kB(As[p], Bs[p], wm, wn, half, r, acc);
    wait_stage();
  }

  float* Yout = Y + (size_t)b * OC_ * L_ + n0;
  #pragma unroll
  for (int mi = 0; mi < 2; ++mi)
    #pragma unroll
    for (int v = 0; v < 8; ++v) {
      const int row = wm * 32 + mi * 16 + v + 8 * half;
      #pragma unroll
      for (int ni = 0; ni < 4; ++ni)
        Yout[(size_t)row * L_ + wn * 64 + ni * 16 + r] = acc[mi][ni][v];
    }
}

// ---------------- Kernel 4: W_y = W_w @ y + W_b, fused BN partial stats -----
__global__ __launch_bounds__(256) void k_W(const float* __restrict__ Y,
                                           const float* __restrict__ Ww,
                                           const float* __restrict__ Wb,
                                           float* __restrict__ WY,
                                           float* __restrict__ SUM,
                                           float* __restrict__ SUMSQ) {
  __shared__ float As[2][128 * LDS_STRIDE];
  __shared__ float Bs[2][KC * BSTR];
  __shared__ float s_sum[128];
  __shared__ float s_sq[128];
  const int tid = threadIdx.x;
  const int b = blockIdx.z;
  const int m0 = blockIdx.y * 128;
  const size_t n0 = (size_t)blockIdx.x * 128;
  if (tid < 128) { s_sum[tid] = 0.f; s_sq[tid] = 0.f; }
  const float* A  = Ww + (size_t)m0 * OC_;
  const float* Yb = Y + (size_t)b * OC_ * L_ + n0;

  v8f acc[2][4] = {};
  const int wave = tid >> 5, lane = tid & 31;
  const int wm = wave >> 1, wn = wave & 1, half = lane >> 4, r = lane & 15;

  stage_A(A, OC_, As[0], tid);
  stage_Bk(Yb, L_, Bs[0], tid);
  wait_stage();
  int p = 0;
  for (int kc = 0; kc < OC_; kc += KC, p ^= 1) {
    if (kc + KC < OC_) {
      stage_A(A + kc + KC, OC_, As[p ^ 1], tid);
      stage_Bk(Yb + (size_t)(kc + KC) * L_, L_, Bs[p ^ 1], tid);
    }
    wave_tile_kB(As[p], Bs[p], wm, wn, half, r, acc);
    wait_stage();
  }

  float* O = WY + ((size_t)b * C_ + m0) * L_ + n0;
  #pragma unroll
  for (int mi = 0; mi < 2; ++mi)
    #pragma unroll
    for (int v = 0; v < 8; ++v) {
      const int row = wm * 32 + mi * 16 + v + 8 * half;
      const float bv = Wb[m0 + row];
      float ps = 0.f, ps2 = 0.f;
      #pragma unroll
      for (int ni = 0; ni < 4; ++ni) {
        const float val = acc[mi][ni][v] + bv;
        O[(size_t)row * L_ + wn * 64 + ni * 16 + r] = val;
        ps += val; ps2 += val * val;
      }
      #pragma unroll
      for (int off = 1; off < 16; off <<= 1) {
        ps  += __shfl_xor(ps,  off, 32);
        ps2 += __shfl_xor(ps2, off, 32);
      }
      if (r == 0) { atomicAdd(&s_sum[row], ps); atomicAdd(&s_sq[row], ps2); }
    }
  __syncthreads();
  if (tid < 128) {
    atomicAdd(&SUM[m0 + tid],   s_sum[tid]);
    atomicAdd(&SUMSQ[m0 + tid], s_sq[tid]);
  }
}

// ---------------- Kernel 5: finalize BN statistics --------------------------
__global__ void k_stats(const float* __restrict__ SUM, const float* __restrict__ SUMSQ,
                        float* __restrict__ MEAN, float* __restrict__ RSTD) {
  const int c = threadIdx.x;
  const float n = (float)(B_ * L_);
  const float m = SUM[c] / n;
  const float var = SUMSQ[c] / n - m * m;
  MEAN[c] = m;
  RSTD[c] = rsqrtf(var + BN_EPS);
}

// ---------------- Kernel 6: normalize + affine + residual -------------------
__global__ __launch_bounds__(256) void k_out(const float* __restrict__ WY,
                                             const float* __restrict__ x,
                                             const float* __restrict__ gamma,
                                             const float* __restrict__ beta,
                                             const float* __restrict__ MEAN,
                                             const float* __restrict__ RSTD,
                                             float* __restrict__ out) {
  const size_t q = (size_t)blockIdx.x * 256 + threadIdx.x;   // float4 index
  const int c = (int)((q >> 10) & (C_ - 1));                 // (q*4 / L) % C
  const float m = MEAN[c], rs = RSTD[c], ga = gamma[c], be = beta[c];
  const float4 w  = ((const float4*)WY)[q];
  const float4 xv = ((const float4*)x)[q];
  float4 o;
  o.x = ga * (w.x - m) * rs + be + xv.x;
  o.y = ga * (w.y - m) * rs + be + xv.y;
  o.z = ga * (w.z - m) * rs + be + xv.z;
  o.w = ga * (w.w - m) * rs + be + xv.w;
  ((float4*)out)[q] = o;
}

__global__ void k_zero(float* __restrict__ p, int n) {
  const int i = blockIdx.x * 256 + threadIdx.x;
  if (i < n) p[i] = 0.f;
}

// ---------------------------------------------------------------------------
extern "C" void kernel_launch(void* const* d_in, const int* in_sizes, int n_in,
                              void* d_out, int out_size, void* d_ws, size_t ws_size,
                              hipStream_t stream) {
  const float* x     = (const float*)d_in[0];
  const float* g_w   = (const float*)d_in[1];
  const float* g_b   = (const float*)d_in[2];
  const float* th_w  = (const float*)d_in[3];
  const float* th_b  = (const float*)d_in[4];
  const float* ph_w  = (const float*)d_in[5];
  const float* ph_b  = (const float*)d_in[6];
  const float* W_w   = (const float*)d_in[7];
  const float* W_b   = (const float*)d_in[8];
  const float* gamma = (const float*)d_in[9];
  const float* beta  = (const float*)d_in[10];
  float* out = (float*)d_out;

  float* ws = (float*)d_ws;
  float* P     = ws;                       // (B,384,L) = 12,582,912 floats
  float* WY    = ws;                       // aliases P (P dead before k_W writes)
  float* S     = ws + (size_t)12582912;    // (B,128,128) = 131,072
  float* Y     = S + 131072;               // (B,128,L)   = 4,194,304
  float* SUM   = Y + 4194304;              // 256
  float* SUMSQ = SUM + 256;                // 256
  float* MEAN  = SUMSQ + 256;              // 256
  float* RSTD  = MEAN + 256;               // 256

  k_zero <<<dim3(512), dim3(256), 0, stream>>>(S, 131072);
  k_zero <<<dim3(4),   dim3(256), 0, stream>>>(SUM, 1024);   // SUM..RSTD
  k_proj <<<dim3(32, 3, 8), dim3(256), 0, stream>>>(x, g_w, g_b, th_w, th_b,
                                                    ph_w, ph_b, P);
  k_S    <<<dim3(16, 8),    dim3(256), 0, stream>>>(P, S);
  k_Y    <<<dim3(32, 8),    dim3(256), 0, stream>>>(P, S, Y);
  k_W    <<<dim3(32, 2, 8), dim3(256), 0, stream>>>(Y, W_w, W_b, WY, SUM, SUMSQ);
  k_stats<<<dim3(1),        dim3(256), 0, stream>>>(SUM, SUMSQ, MEAN, RSTD);
  k_out  <<<dim3(8192),     dim3(256), 0, stream>>>(WY, x, gamma, beta, MEAN,
                                                    RSTD, out);
}